// contResLoss_88888643158594
// MI455X (gfx1250) — compile-verified
//
#include <hip/hip_runtime.h>

typedef __attribute__((ext_vector_type(2))) float v2f;
typedef __attribute__((ext_vector_type(8))) float v8f;

#define N_NODES 16384
#define N_EDGES 524288
#define TILE    8192   // floats of w staged in LDS (32 KB)

// ---------------------------------------------------------------------------
// Phase A: w[i] = V[i,0] + V[i,1]; zero the scatter targets and the output.
// ---------------------------------------------------------------------------
__global__ void k_init(const float* __restrict__ V, float* __restrict__ w,
                       float* __restrict__ s, float* __restrict__ dcnt,
                       float* __restrict__ out) {
    int i = blockIdx.x * blockDim.x + threadIdx.x;
    if (i < N_NODES) {
        float2 v = ((const float2*)V)[i];
        w[i]    = v.x + v.y;
        s[i]    = 0.0f;
        dcnt[i] = 0.0f;
    }
    if (i == 0) out[0] = 0.0f;
}

// ---------------------------------------------------------------------------
// Phase B: s[dst] += w[src]; d[dst] += 1  (edge scatter, L2-resident targets)
// ---------------------------------------------------------------------------
__global__ void k_scatter(const int* __restrict__ edges,
                          const float* __restrict__ w,
                          float* __restrict__ s, float* __restrict__ dcnt) {
    int e = blockIdx.x * blockDim.x + threadIdx.x;
    if (e >= N_EDGES) return;
    int src = edges[e];            // edges row 0
    int dst = edges[N_EDGES + e];  // edges row 1
    atomicAdd(&s[dst], w[src]);
    atomicAdd(&dcnt[dst], 1.0f);
}

// ---------------------------------------------------------------------------
// Phase C: total = sum_{i,k} |s_k - w_i * d_k|, reduced via f32 WMMA with a
// ones A-matrix:  D = ones(16x4) x B(4x16) + C  => column sums accumulate in C.
// One wave owns a 16-column strip of k; i is swept 16 rows per step using 4
// independent accumulators to break the WMMA RAW chain.
// ---------------------------------------------------------------------------
__global__ void k_reduce(const float* __restrict__ w,
                         const float* __restrict__ s,
                         const float* __restrict__ dcnt,
                         float* __restrict__ out) {
    __shared__ float lw[TILE];

    const int tid   = threadIdx.x;
    const int lane  = tid & 31;
    const int wave  = tid >> 5;          // 8 waves per block
    const int half  = lane >> 4;         // 0: B rows K=0,1   1: B rows K=2,3
    const int col   = lane & 15;         // B column N
    const int strip = blockIdx.x * 8 + wave;
    const int k     = strip * 16 + col;

    const float sk = s[k];
    const float dk = dcnt[k];

    v8f acc0 = {0.f,0.f,0.f,0.f,0.f,0.f,0.f,0.f};
    v8f acc1 = acc0, acc2 = acc0, acc3 = acc0;
    v2f ones; ones[0] = 1.0f; ones[1] = 1.0f;

    for (int t = 0; t < N_NODES; t += TILE) {
        // cooperative stage of w[t .. t+TILE) into LDS (float4 per thread)
        const float4* src4 = (const float4*)(w + t);
        float4*       dst4 = (float4*)lw;
#pragma unroll
        for (int j = 0; j < TILE / 4 / 256; ++j)
            dst4[tid + j * 256] = src4[tid + j * 256];
        __syncthreads();

        for (int i = 0; i < TILE; i += 16) {
            // 4 WMMAs, 4 rows each, independent accumulators
            {
                float2 wp = *(const float2*)&lw[i + 0  + 2 * half];
                v2f b; b[0] = fabsf(sk - wp.x * dk); b[1] = fabsf(sk - wp.y * dk);
                acc0 = __builtin_amdgcn_wmma_f32_16x16x4_f32(
                           false, ones, false, b, (short)0, acc0, false, false);
            }
            {
                float2 wp = *(const float2*)&lw[i + 4  + 2 * half];
                v2f b; b[0] = fabsf(sk - wp.x * dk); b[1] = fabsf(sk - wp.y * dk);
                acc1 = __builtin_amdgcn_wmma_f32_16x16x4_f32(
                           false, ones, false, b, (short)0, acc1, false, false);
            }
            {
                float2 wp = *(const float2*)&lw[i + 8  + 2 * half];
                v2f b; b[0] = fabsf(sk - wp.x * dk); b[1] = fabsf(sk - wp.y * dk);
                acc2 = __builtin_amdgcn_wmma_f32_16x16x4_f32(
                           false, ones, false, b, (short)0, acc2, false, false);
            }
            {
                float2 wp = *(const float2*)&lw[i + 12 + 2 * half];
                v2f b; b[0] = fabsf(sk - wp.x * dk); b[1] = fabsf(sk - wp.y * dk);
                acc3 = __builtin_amdgcn_wmma_f32_16x16x4_f32(
                           false, ones, false, b, (short)0, acc3, false, false);
            }
        }
        __syncthreads();
    }

    // With A == ones, every row of D carries the same column sum.
    // Lanes 0..15, element 0 hold the totals for columns k0..k0+15.
    v8f accA = acc0 + acc1;
    v8f accB = acc2 + acc3;
    v8f acc  = accA + accB;

    const float SCALE = 0.05f / (16384.0f * 16384.0f);
    if (lane < 16) atomicAdd(out, acc[0] * SCALE);
}

// ---------------------------------------------------------------------------
extern "C" void kernel_launch(void* const* d_in, const int* in_sizes, int n_in,
                              void* d_out, int out_size, void* d_ws, size_t ws_size,
                              hipStream_t stream) {
    (void)in_sizes; (void)n_in; (void)out_size; (void)ws_size;

    const float* V     = (const float*)d_in[0];
    const int*   edges = (const int*)d_in[1];
    float*       out   = (float*)d_out;

    float* w    = (float*)d_ws;      // N floats
    float* s    = w + N_NODES;       // N floats
    float* dcnt = s + N_NODES;       // N floats

    k_init   <<<N_NODES / 256, 256, 0, stream>>>(V, w, s, dcnt, out);
    k_scatter<<<N_EDGES / 256, 256, 0, stream>>>(edges, w, s, dcnt);
    // 1024 strips of 16 columns, 8 waves (256 threads) per block -> 128 blocks
    k_reduce <<<(N_NODES / 16) / 8, 256, 0, stream>>>(w, s, dcnt, out);
}